// EGHG_13134009991424
// MI455X (gfx1250) — compile-verified
//
#include <hip/hip_runtime.h>

#define NUM_USERS 100000
#define NUM_ITEMS 50000
#define N_NODES   150000
#define DIM       64
#define N_EDGES   1200000
#define KB        0.8f
#define OMK       0.2f   // (1 - K)

typedef __attribute__((ext_vector_type(2))) float v2f;
typedef __attribute__((ext_vector_type(8))) float v8f;

static constexpr int TOTAL4 = N_NODES * DIM / 4;     // float4 elements per full array
static constexpr int UCNT4  = NUM_USERS * DIM / 4;   // float4 elements of user part

// acc = A = concat(user_emb, item_emb); B = (1-K)*A   (pre-scale for layer 1)
__global__ void lgcn_init(const float4* __restrict__ ue, const float4* __restrict__ ie,
                          float4* __restrict__ acc, float4* __restrict__ embA,
                          float4* __restrict__ embB) {
    int i = blockIdx.x * blockDim.x + threadIdx.x;
    if (i >= TOTAL4) return;
    float4 v = (i < UCNT4) ? ue[i] : ie[i - UCNT4];
    acc[i]  = v;
    embA[i] = v;
    float4 s; s.x = OMK * v.x; s.y = OMK * v.y; s.z = OMK * v.z; s.w = OMK * v.w;
    embB[i] = s;
}

// 16 lanes per edge (2 edges per wave32): y[row] += K*val * x[col].
// Each lane: one global_load_b128 gather + 4 hardware global_atomic_add_f32
// (forced via inline asm; offsets 0/4/8/12 off one base VGPR-pair address).
__global__ void lgcn_scatter(const int* __restrict__ rows, const int* __restrict__ cols,
                             const float* __restrict__ vals,
                             const float* __restrict__ x, float* __restrict__ y) {
    int gtid = blockIdx.x * blockDim.x + threadIdx.x;
    int e    = gtid >> 4;
    int lane = gtid & 15;
    if (e >= N_EDGES) return;
    int   r = rows[e];
    int   c = cols[e];
    float w = KB * vals[e];
    const float4* xp = (const float4*)(x + (size_t)c * DIM);
    float4 m = xp[lane];
    float* yp = y + (size_t)r * DIM + lane * 4;
    asm volatile(
        "global_atomic_add_f32 %0, %1, off\n\t"
        "global_atomic_add_f32 %0, %2, off offset:4\n\t"
        "global_atomic_add_f32 %0, %3, off offset:8\n\t"
        "global_atomic_add_f32 %0, %4, off offset:12"
        :
        : "v"(yp), "v"(w * m.x), "v"(w * m.y), "v"(w * m.z), "v"(w * m.w)
        : "memory");
}

// acc += res; optionally nxt = (1-K)*res  (pre-scale destination of next layer)
__global__ void lgcn_post(const float4* __restrict__ res, float4* __restrict__ acc,
                          float4* __restrict__ nxt, int writeScale) {
    int i = blockIdx.x * blockDim.x + threadIdx.x;
    if (i >= TOTAL4) return;
    float4 t = res[i];
    float4 a = acc[i];
    a.x += t.x; a.y += t.y; a.z += t.z; a.w += t.w;
    acc[i] = a;
    if (writeScale) {
        float4 s; s.x = OMK * t.x; s.y = OMK * t.y; s.z = OMK * t.z; s.w = OMK * t.w;
        nxt[i] = s;
    }
}

// Final: gamma[b] = dot(acc_u[users[b]], acc_i[items[b]]) / 16  via V_WMMA_F32_16X16X4_F32.
// 16 pairs per wave; diagonal of (U 16x64) x (I^T 64x16) accumulated in 16 K=4 steps.
// A layout (16x4 f32): lanes 0-15 -> K=0,1 in v[0],v[1]; lanes 16-31 -> K=2,3.
// C/D layout: VGPR r: lanes 0-15 -> M=r,N=lane; lanes 16-31 -> M=r+8,N=lane-16.
__global__ void lgcn_gather_dot(const int* __restrict__ users, const int* __restrict__ items,
                                const float* __restrict__ acc, float* __restrict__ out, int B) {
    int wave = (blockIdx.x * blockDim.x + threadIdx.x) >> 5;
    int lane = threadIdx.x & 31;
    int base = wave * 16;
    if (base >= B) return;
    int m  = lane & 15;       // pair (row of A / col of B) handled by this lane
    int hi = lane >> 4;       // selects K sub-pair (0,1) vs (2,3)
    int p  = base + m; if (p >= B) p = B - 1;

    const float* urow = acc + (size_t)users[p] * DIM;
    const float* irow = acc + ((size_t)NUM_USERS + (size_t)items[p]) * DIM;

    v8f c = {};
#pragma unroll
    for (int k0 = 0; k0 < DIM; k0 += 4) {
        int k = k0 + hi * 2;
        v2f a; a.x = urow[k]; a.y = urow[k + 1];
        v2f b; b.x = irow[k]; b.y = irow[k + 1];
        c = __builtin_amdgcn_wmma_f32_16x16x4_f32(false, a, false, b, (short)0, c,
                                                  false, false);
    }
    // Diagonal extraction: lane r (r<8) has gamma[r] in c[r]; lane 24+j has gamma[8+j] in c[j].
    int d = (lane < 8) ? lane : (lane - 24);
    float g = 0.0f;
#pragma unroll
    for (int j = 0; j < 8; ++j) g = (d == j) ? c[j] : g;
    if (lane < 8) {
        int o = base + lane;
        if (o < B) out[o] = g * (1.0f / 16.0f);
    } else if (lane >= 24) {
        int o = base + lane - 16;
        if (o < B) out[o] = g * (1.0f / 16.0f);
    }
}

extern "C" void kernel_launch(void* const* d_in, const int* in_sizes, int n_in,
                              void* d_out, int out_size, void* d_ws, size_t ws_size,
                              hipStream_t stream) {
    const int*   users = (const int*)d_in[0];
    const int*   items = (const int*)d_in[1];
    const int*   erow  = (const int*)d_in[2];
    const int*   ecol  = (const int*)d_in[3];
    const float* eval  = (const float*)d_in[4];
    const float* uemb  = (const float*)d_in[5];
    const float* iemb  = (const float*)d_in[6];
    float* out = (float*)d_out;

    size_t nf = (size_t)N_NODES * DIM;
    float* acc  = (float*)d_ws;
    float* embA = acc + nf;
    float* embB = embA + nf;

    dim3 eb(256), eg((TOTAL4 + 255) / 256);
    lgcn_init<<<eg, eb, 0, stream>>>((const float4*)uemb, (const float4*)iemb,
                                     (float4*)acc, (float4*)embA, (float4*)embB);

    int scatterBlocks = (N_EDGES * 16 + 255) / 256;   // 75,000 blocks

    // layer 1: A -> B (B pre-scaled by init)
    lgcn_scatter<<<scatterBlocks, 256, 0, stream>>>(erow, ecol, eval, embA, embB);
    lgcn_post<<<eg, eb, 0, stream>>>((const float4*)embB, (float4*)acc, (float4*)embA, 1);
    // layer 2: B -> A (A pre-scaled by post 1)
    lgcn_scatter<<<scatterBlocks, 256, 0, stream>>>(erow, ecol, eval, embB, embA);
    lgcn_post<<<eg, eb, 0, stream>>>((const float4*)embA, (float4*)acc, (float4*)embB, 1);
    // layer 3: A -> B (B pre-scaled by post 2)
    lgcn_scatter<<<scatterBlocks, 256, 0, stream>>>(erow, ecol, eval, embA, embB);
    lgcn_post<<<eg, eb, 0, stream>>>((const float4*)embB, (float4*)acc, (float4*)embA, 0);

    int B = in_sizes[0];                       // 4096 pairs
    int waves = (B + 15) / 16;
    int threads = waves * 32;
    int blocks = (threads + 255) / 256;
    lgcn_gather_dot<<<blocks, 256, 0, stream>>>(users, items, acc, out, B);
}